// LearnedSparseScalarObservationFromNeighbors_32100585570823
// MI455X (gfx1250) — compile-verified
//
#include <hip/hip_runtime.h>
#include <hip/hip_bf16.h>

typedef float v2f __attribute__((ext_vector_type(2)));
typedef float v8f __attribute__((ext_vector_type(8)));

#define N_FEAT 192
#define N_LON  720
#define N_LAT  360
#define PLANE  (N_LON * N_LAT)
#define NQ     100000
#define HID    256
#define OUT_D  8
#define IN_PAD 196          // 194 zero-padded to multiple of WMMA K=4
#define ROWW   260          // LDS row stride (floats); 260 mod 64 = 4 -> conflict-free column reads
#define WAVES  4            // waves per block
#define TILE_M 16
#define SLICE  (TILE_M * ROWW)      // one 16-row tile region (floats)

__device__ __forceinline__ float gelu_tanh(float x) {
    // jax.nn.gelu default (approximate=True)
    const float c0 = 0.7978845608028654f;   // sqrt(2/pi)
    float inner = c0 * (x + 0.044715f * x * x * x);
    return 0.5f * x * (1.0f + tanhf(inner));
}

__global__ __launch_bounds__(WAVES * 32)
void LearnedSparseObs_mlp_wmma_kernel(
    const float* __restrict__ features,
    const float* __restrict__ lon_q, const float* __restrict__ lat_q,
    const float* __restrict__ lon_grid, const float* __restrict__ lat_grid,
    const float* __restrict__ W1, const float* __restrict__ b1,
    const float* __restrict__ W2, const float* __restrict__ b2,
    const float* __restrict__ W3, const float* __restrict__ b3,
    float* __restrict__ out)
{
    extern __shared__ float smem[];
    const int lane = threadIdx.x & 31;
    const int wave = threadIdx.x >> 5;
    float* Xl = smem + wave * (2 * SLICE);   // region A: X, later H2
    float* Hl = Xl + SLICE;                  // region B: H1

    const int tile   = blockIdx.x * WAVES + wave;
    const int qbase  = tile * TILE_M;
    const int m      = lane & 15;            // A-row / query row held by this lane
    const int laneN  = lane & 15;            // B/C/D column held by this lane
    const int kk     = (lane >> 4) * 2;      // K sub-offset of A/B fragments
    const int rowOff = (lane >> 4) * 8;      // C/D: VGPR r <-> M = r + rowOff

    // ---- nearest-neighbor indices & displacement features (analytic) ----
    int  q  = qbase + m;
    bool qv = q < NQ;
    int  qc = qv ? q : (NQ - 1);             // clamp for safe loads on tail tiles
    float lon = lon_q[qc];
    float lat = lat_q[qc];
    int li = (int)floorf(lon * 2.0f + 0.5f); // lon grid spacing = 0.5 deg, periodic
    li = (li >= N_LON) ? li - N_LON : (li < 0 ? li + N_LON : li);
    int la = (int)floorf((lat + 90.0f) * (359.0f / 180.0f) + 0.5f);
    la = la < 0 ? 0 : (la > N_LAT - 1 ? N_LAT - 1 : la);
    float dlon = lon - lon_grid[li];         // plain (unwrapped) diff, like reference
    float dlat = lat - lat_grid[la];
    const int gbase = li * N_LAT + la;

    // ---- gather 192 features into LDS X tile [16 x 196] (stride ROWW) ----
    {
        const int f0 = lane >> 4;            // lanes 0-15: even f, 16-31: odd f
        for (int f = f0; f < N_FEAT; f += 2)
            Xl[m * ROWW + f] = features[f * PLANE + gbase];
        if (lane < 16) {
            Xl[m * ROWW + 192] = dlon;
            Xl[m * ROWW + 193] = dlat;
            Xl[m * ROWW + 194] = 0.0f;       // K padding
            Xl[m * ROWW + 195] = 0.0f;
        }
    }

    v8f acc[8];                               // 64 VGPRs of accumulators (per group)

    // ========== layer 1: X[16x196] @ W1[194x256] + b1, gelu -> H1 ==========
    for (int g = 0; g < 2; ++g) {
        const int n0 = g * 128;
#pragma unroll
        for (int nt = 0; nt < 8; ++nt) {
            float bv = b1[n0 + nt * 16 + laneN];
#pragma unroll
            for (int r = 0; r < 8; ++r) acc[nt][r] = bv;
        }
#pragma unroll 2
        for (int k = 0; k < IN_PAD; k += 4) {
            v2f a = *(const v2f*)&Xl[m * ROWW + k + kk];   // 8B LDS load
            int krow = k + kk;
            if (krow > 192) krow = 192;       // pad rows: A is zero there anyway
            const float* wk = W1 + krow * HID + n0 + laneN;
#pragma unroll
            for (int nt = 0; nt < 8; ++nt) {
                v2f b;
                b.x = wk[nt * 16];
                b.y = wk[HID + nt * 16];
                acc[nt] = __builtin_amdgcn_wmma_f32_16x16x4_f32(
                    false, a, false, b, (short)0, acc[nt], false, false);
            }
        }
        // write to H region (not read by layer 1) -> no hazard, accs retire now
#pragma unroll
        for (int nt = 0; nt < 8; ++nt)
#pragma unroll
            for (int r = 0; r < 8; ++r)
                Hl[(rowOff + r) * ROWW + n0 + nt * 16 + laneN] = gelu_tanh(acc[nt][r]);
    }

    // ========== layer 2: H1[16x256] @ W2[256x256] + b2, gelu -> X (H2) ==========
    for (int g = 0; g < 2; ++g) {
        const int n0 = g * 128;
#pragma unroll
        for (int nt = 0; nt < 8; ++nt) {
            float bv = b2[n0 + nt * 16 + laneN];
#pragma unroll
            for (int r = 0; r < 8; ++r) acc[nt][r] = bv;
        }
#pragma unroll 2
        for (int k = 0; k < HID; k += 4) {
            v2f a = *(const v2f*)&Hl[m * ROWW + k + kk];
            const float* wk = W2 + (k + kk) * HID + n0 + laneN;
#pragma unroll
            for (int nt = 0; nt < 8; ++nt) {
                v2f b;
                b.x = wk[nt * 16];
                b.y = wk[HID + nt * 16];
                acc[nt] = __builtin_amdgcn_wmma_f32_16x16x4_f32(
                    false, a, false, b, (short)0, acc[nt], false, false);
            }
        }
        // write H2 into the X region (features no longer needed)
#pragma unroll
        for (int nt = 0; nt < 8; ++nt)
#pragma unroll
            for (int r = 0; r < 8; ++r)
                Xl[(rowOff + r) * ROWW + n0 + nt * 16 + laneN] = gelu_tanh(acc[nt][r]);
    }

    // ========== layer 3: H2[16x256] @ W3[256x8] + b3 (N padded to 16) ==========
    const float nmask = (laneN < OUT_D) ? 1.0f : 0.0f;
    const int   nc    = laneN & (OUT_D - 1);
    v8f acc3;
    {
        float bv = b3[nc] * nmask;
#pragma unroll
        for (int r = 0; r < 8; ++r) acc3[r] = bv;
    }
#pragma unroll 2
    for (int k = 0; k < HID; k += 4) {
        v2f a = *(const v2f*)&Xl[m * ROWW + k + kk];
        v2f b;
        b.x = W3[(k + kk) * OUT_D + nc] * nmask;   // masked pad columns, in-bounds loads
        b.y = W3[(k + kk + 1) * OUT_D + nc] * nmask;
        acc3 = __builtin_amdgcn_wmma_f32_16x16x4_f32(
            false, a, false, b, (short)0, acc3, false, false);
    }

    // ---- store D: lane holds column laneN for rows rowOff..rowOff+7 ----
    if (laneN < OUT_D) {
#pragma unroll
        for (int r = 0; r < 8; ++r) {
            int qq = qbase + rowOff + r;
            if (qq < NQ) out[qq * OUT_D + laneN] = acc3[r];
        }
    }
}

extern "C" void kernel_launch(void* const* d_in, const int* in_sizes, int n_in,
                              void* d_out, int out_size, void* d_ws, size_t ws_size,
                              hipStream_t stream) {
    const float* features = (const float*)d_in[0];
    const float* lon_q    = (const float*)d_in[1];
    const float* lat_q    = (const float*)d_in[2];
    const float* lon_grid = (const float*)d_in[3];
    const float* lat_grid = (const float*)d_in[4];
    const float* W1       = (const float*)d_in[5];
    const float* b1       = (const float*)d_in[6];
    const float* W2       = (const float*)d_in[7];
    const float* b2       = (const float*)d_in[8];
    const float* W3       = (const float*)d_in[9];
    const float* b3       = (const float*)d_in[10];
    float* out = (float*)d_out;

    const int tiles  = (NQ + TILE_M - 1) / TILE_M;       // 6250
    const int blocks = (tiles + WAVES - 1) / WAVES;      // 1563
    const size_t shmem = (size_t)WAVES * 2 * SLICE * sizeof(float); // 133,120 B < 320 KB

    LearnedSparseObs_mlp_wmma_kernel<<<blocks, WAVES * 32, shmem, stream>>>(
        features, lon_q, lat_q, lon_grid, lat_grid,
        W1, b1, W2, b2, W3, b3, out);
}